// DecodeStepModule_72997264163428
// MI455X (gfx1250) — compile-verified
//
#include <hip/hip_runtime.h>
#include <hip/hip_bf16.h>
#include <math.h>

// ---------------------------------------------------------------------------
// Shapes (from reference): H=2048, NUM_K=16, NUM_V=32, DK=DV=128, KSZ=4
// KEY_DIM=2048, VALUE_DIM=4096, CONV_DIM=8192, VPK=2
// Outputs (flat, in order): hidden_out[2048] ++ new_conv_state[8192*4] ++ rec[32*128*128]
// ---------------------------------------------------------------------------

typedef float v2f __attribute__((ext_vector_type(2)));
typedef float v8f __attribute__((ext_vector_type(8)));

__device__ __forceinline__ float waveReduceSum(float v) {
    // wave32 butterfly
    #pragma unroll
    for (int o = 16; o > 0; o >>= 1) v += __shfl_xor(v, o, 32);
    return v;
}

// ---------------------------------------------------------------------------
// Kernel 1: fused projection GEMVs.  rows: [0,8192)=qkv, [8192,12288)=z,
// [12288,12320)=b, [12320,12352)=a.  One wave per 2048-long row.
// Coalesced float4 streaming of the weight rows (bandwidth-critical path).
// ---------------------------------------------------------------------------
__global__ __launch_bounds__(256) void projKernel(
    const float* __restrict__ h,
    const float* __restrict__ qkv_w, const float* __restrict__ z_w,
    const float* __restrict__ b_w,   const float* __restrict__ a_w,
    float* __restrict__ oq, float* __restrict__ oz,
    float* __restrict__ ob, float* __restrict__ oa)
{
    const int row  = blockIdx.x * 8 + (threadIdx.x >> 5);
    const int lane = threadIdx.x & 31;
    const float* w; float* dst;
    if (row < 8192)       { w = qkv_w + (size_t)row * 2048;          dst = oq + row;           }
    else if (row < 12288) { w = z_w   + (size_t)(row - 8192) * 2048; dst = oz + (row - 8192);  }
    else if (row < 12320) { w = b_w   + (size_t)(row - 12288) * 2048; dst = ob + (row - 12288); }
    else                  { w = a_w   + (size_t)(row - 12320) * 2048; dst = oa + (row - 12320); }

    const float4* w4 = (const float4*)w;
    const float4* h4 = (const float4*)h;
    float acc = 0.f;
    #pragma unroll 4
    for (int i = lane; i < 512; i += 32) {
        float4 a = w4[i], b = h4[i];
        acc += a.x * b.x + a.y * b.y + a.z * b.z + a.w * b.w;
    }
    acc = waveReduceSum(acc);
    if (lane == 0) *dst = acc;
}

// ---------------------------------------------------------------------------
// Kernel 2: conv-state shift + depthwise conv + SiLU; q/k L2-norm; beta & g.
// 64 blocks x 128 threads; blocks 0..15 -> q heads, 16..31 -> k heads.
// ---------------------------------------------------------------------------
__global__ __launch_bounds__(128) void convKernel(
    const float* __restrict__ qkv, const float* __restrict__ conv_state,
    const float* __restrict__ conv_w,
    const float* __restrict__ bvec, const float* __restrict__ avec,
    const float* __restrict__ A_log, const float* __restrict__ dt_bias,
    float* __restrict__ ncs_out, float* __restrict__ conv_out,
    float* __restrict__ qn, float* __restrict__ kn,
    float* __restrict__ beta, float* __restrict__ gexp)
{
    __shared__ float red[4];
    const int c = blockIdx.x * 128 + threadIdx.x;

    float4 cs = ((const float4*)conv_state)[c];
    float4 w  = ((const float4*)conv_w)[c];
    float  x  = qkv[c];
    float4 ncs = make_float4(cs.y, cs.z, cs.w, x);
    ((float4*)ncs_out)[c] = ncs;                       // new_conv_state

    float acc = ncs.x * w.x + ncs.y * w.y + ncs.z * w.z + ncs.w * w.w;
    float co  = acc / (1.f + __expf(-acc));            // SiLU
    conv_out[c] = co;

    if (blockIdx.x < 32) {                             // uniform per block
        float s = waveReduceSum(co * co);
        if ((threadIdx.x & 31) == 0) red[threadIdx.x >> 5] = s;
        __syncthreads();
        float tot   = red[0] + red[1] + red[2] + red[3];
        float scale = rsqrtf(tot + 1e-6f);
        if (blockIdx.x < 16) qn[c] = co * scale * 0.08838834764831845f; // * 1/sqrt(128)
        else                 kn[c - 2048] = co * scale;
    }

    if (blockIdx.x == 63 && threadIdx.x < 32) {        // tiny scalar block
        int n = threadIdx.x;
        beta[n] = 1.f / (1.f + __expf(-bvec[n]));
        float xx = avec[n] + dt_bias[n];
        float sp = (xx > 20.f) ? xx : log1pf(__expf(xx));  // softplus
        gexp[n]  = __expf(-__expf(A_log[n]) * sp);
    }
}

// ---------------------------------------------------------------------------
// Kernel 3: per-head recurrent state update using V_WMMA_F32_16X16X4_F32.
// 32 blocks (one head), 256 threads = 8 waves; wave w owns v-tile [16w,16w+16).
// WMMA computes D = A x B with A[i][kk] = rec[k0+kk][v0+i] (transposed read of
// the k-major state) and B[kk][*] = broadcast of k-hat (resp. q-hat), so the
// two accumulator chains yield t_k[v] = rec^T k and t_q[v] = rec^T q with the
// A operand shared (classic reuse pattern).  Then:
//   delta[v] = (v_in[v] - g*t_k[v]) * beta
//   core[v]  = g*t_q[v] + (k.q) * delta[v]          (rec_new^T q, algebraically)
//   rec_new  = g*rec + outer(k, delta)              (coalesced second pass)
// ---------------------------------------------------------------------------
__global__ __launch_bounds__(256) void headUpdateKernel(
    const float* __restrict__ rec_in, const float* __restrict__ qn,
    const float* __restrict__ kn,     const float* __restrict__ conv_out,
    const float* __restrict__ beta_arr, const float* __restrict__ gexp_arr,
    float* __restrict__ rec_out, float* __restrict__ core)
{
    __shared__ float s_k[128], s_q[128], s_delta[128];
    const int hi   = blockIdx.x;
    const int tid  = threadIdx.x;
    const int kidx = hi >> 1;                     // VPK=2 head repeat

    if (tid < 128) {
        s_k[tid] = kn[kidx * 128 + tid];
        s_q[tid] = qn[kidx * 128 + tid];
    }
    __syncthreads();

    const float gexp = gexp_arr[hi];
    const float beta = beta_arr[hi];
    const int wid  = tid >> 5, lane = tid & 31;
    const int half = lane >> 4, lrow = lane & 15;
    const int v0   = wid * 16;
    const int kb   = 2 * half;
    const float* recH = rec_in + (size_t)hi * 16384;

    // k.q dot, redundant per wave (cheap, keeps EXEC uniform for WMMA)
    float qk = 0.f;
    for (int i = lane; i < 128; i += 32) qk += s_k[i] * s_q[i];
    qk = waveReduceSum(qk);

    v8f ck = {0.f,0.f,0.f,0.f,0.f,0.f,0.f,0.f};
    v8f cq = {0.f,0.f,0.f,0.f,0.f,0.f,0.f,0.f};
    for (int k0 = 0; k0 < 128; k0 += 4) {
        v2f A, Bk, Bq;
        // A 16x4 layout: lanes 0-15 hold K=0,1 in v0,v1; lanes 16-31 hold K=2,3
        A.x = recH[(size_t)(k0 + kb)     * 128 + v0 + lrow];
        A.y = recH[(size_t)(k0 + kb + 1) * 128 + v0 + lrow];
        // B rows broadcast the vector value for their K index
        Bk.x = s_k[k0 + kb];  Bk.y = s_k[k0 + kb + 1];
        Bq.x = s_q[k0 + kb];  Bq.y = s_q[k0 + kb + 1];
        ck = __builtin_amdgcn_wmma_f32_16x16x4_f32(false, A, false, Bk, (short)0, ck, false, false);
        cq = __builtin_amdgcn_wmma_f32_16x16x4_f32(false, A, false, Bq, (short)0, cq, false, false);
    }

    // C/D layout: lane L, reg r -> M = r + 8*(L>>4), N = L&15; columns replicated
    if (lrow == 0) {
        #pragma unroll
        for (int r = 0; r < 8; ++r) {
            int v   = v0 + 8 * half + r;
            float kv = gexp * ck[r];
            float d  = (conv_out[4096 + hi * 128 + v] - kv) * beta;
            s_delta[v] = d;
            core[hi * 128 + v] = gexp * cq[r] + qk * d;
        }
    }
    __syncthreads();

    // rec_new = g*rec + k (x) delta   — fully coalesced stream
    float* recO = rec_out + (size_t)hi * 16384;
    for (int idx = tid; idx < 16384; idx += 256) {
        int k = idx >> 7, v = idx & 127;
        recO[idx] = gexp * recH[idx] + s_k[k] * s_delta[v];
    }
}

// ---------------------------------------------------------------------------
// Kernel 4: RMS-norm (DV=128) * norm_w, gated by SiLU(z).
// ---------------------------------------------------------------------------
__global__ __launch_bounds__(128) void normGateKernel(
    const float* __restrict__ core, const float* __restrict__ z,
    const float* __restrict__ norm_w, float* __restrict__ outg)
{
    __shared__ float red[4];
    const int hi = blockIdx.x, t = threadIdx.x;
    float x = core[hi * 128 + t];
    float s = waveReduceSum(x * x);
    if ((threadIdx.x & 31) == 0) red[threadIdx.x >> 5] = s;
    __syncthreads();
    float var = (red[0] + red[1] + red[2] + red[3]) * (1.f / 128.f);
    float xn  = x * rsqrtf(var + 1e-6f) * norm_w[t];
    float zv  = z[hi * 128 + t];
    outg[hi * 128 + t] = xn * (zv / (1.f + __expf(-zv)));
}

// ---------------------------------------------------------------------------
// Kernel 5: out_proj GEMV, 2048 rows x 4096, one wave per row.
// ---------------------------------------------------------------------------
__global__ __launch_bounds__(256) void outProjKernel(
    const float* __restrict__ og, const float* __restrict__ w,
    float* __restrict__ hid)
{
    const int row  = blockIdx.x * 8 + (threadIdx.x >> 5);
    const int lane = threadIdx.x & 31;
    const float4* w4 = (const float4*)(w + (size_t)row * 4096);
    const float4* g4 = (const float4*)og;
    float acc = 0.f;
    #pragma unroll 4
    for (int i = lane; i < 1024; i += 32) {
        float4 a = w4[i], b = g4[i];
        acc += a.x * b.x + a.y * b.y + a.z * b.z + a.w * b.w;
    }
    acc = waveReduceSum(acc);
    if (lane == 0) hid[row] = acc;
}

// ---------------------------------------------------------------------------
extern "C" void kernel_launch(void* const* d_in, const int* in_sizes, int n_in,
                              void* d_out, int out_size, void* d_ws, size_t ws_size,
                              hipStream_t stream) {
    const float* h          = (const float*)d_in[0];
    const float* conv_state = (const float*)d_in[1];
    const float* rec_state  = (const float*)d_in[2];
    const float* conv_w     = (const float*)d_in[3];
    const float* qkv_w      = (const float*)d_in[4];
    const float* z_w        = (const float*)d_in[5];
    const float* b_w        = (const float*)d_in[6];
    const float* a_w        = (const float*)d_in[7];
    const float* out_w      = (const float*)d_in[8];
    const float* dt_bias    = (const float*)d_in[9];
    const float* A_log      = (const float*)d_in[10];
    const float* norm_w     = (const float*)d_in[11];

    float* out        = (float*)d_out;
    float* hidden_out = out;                    // 2048
    float* ncs_out    = out + 2048;             // 8192*4
    float* rec_out    = out + 2048 + 32768;     // 32*128*128

    float* ws      = (float*)d_ws;
    float* ws_qkv  = ws;                        // 8192
    float* ws_z    = ws + 8192;                 // 4096
    float* ws_b    = ws + 12288;                // 32
    float* ws_a    = ws + 12320;                // 32
    float* ws_co   = ws + 12352;                // 8192 (conv_out)
    float* ws_qn   = ws + 20544;                // 2048
    float* ws_kn   = ws + 22592;                // 2048
    float* ws_beta = ws + 24640;                // 32
    float* ws_gexp = ws + 24672;                // 32
    float* ws_core = ws + 24704;                // 4096
    float* ws_outg = ws + 28800;                // 4096

    projKernel<<<1544, 256, 0, stream>>>(h, qkv_w, z_w, b_w, a_w,
                                         ws_qkv, ws_z, ws_b, ws_a);
    convKernel<<<64, 128, 0, stream>>>(ws_qkv, conv_state, conv_w,
                                       ws_b, ws_a, A_log, dt_bias,
                                       ncs_out, ws_co, ws_qn, ws_kn,
                                       ws_beta, ws_gexp);
    headUpdateKernel<<<32, 256, 0, stream>>>(rec_state, ws_qn, ws_kn, ws_co,
                                             ws_beta, ws_gexp, rec_out, ws_core);
    normGateKernel<<<32, 128, 0, stream>>>(ws_core, ws_z, norm_w, ws_outg);
    outProjKernel<<<256, 256, 0, stream>>>(ws_outg, out_w, hidden_out);
}